// LDDMMShootingScalarMomentumNet_32427003085494
// MI455X (gfx1250) — compile-verified
//
#include <hip/hip_runtime.h>

// ---------------------------------------------------------------------------
// LDDMM shooting (EPDiff scalar momentum), B=8, 512x512, 10 RK4 steps.
// Gaussian FFT smoother replaced by exact circulant GEMMs: smooth(X)=G*X*G,
// executed with CDNA5 V_WMMA_F32_16X16X4_F32 (fp32 matrix path).
// B tile is stored transposed in LDS so both A and B fragments are single
// aligned ds_load_b64 ops feeding the WMMA directly (no VGPR shuffles).
// ---------------------------------------------------------------------------

typedef float v2f __attribute__((ext_vector_type(2)));
typedef float v8f __attribute__((ext_vector_type(8)));

#define HN 512
#define WN 512
#define BATCH 8
#define HW (HN * WN)
#define FIELD (BATCH * HW)        // floats per field (8 * 512 * 512)
#define RKSTEPS 10
#define DT 0.1f
#define INV_H 511.0f              // 1 / grid spacing

// GEMM tiling: 256 threads = 8 wave32; block tile 128(M) x 64(N), K chunk 64.
#define BM 128
#define BN 64
#define BK 64
#define LDA 68    // sA[m][k]: frag read banks (4m+k)%64 -> lane halves disjoint mod 4
#define LDBT 68   // sBt[n][k] transposed: frag read banks (4n+k)%64 -> conflict-free

// ---------------------------------------------------------------------------
// g = irfft( exp(-0.5*sigma^2*k^2) ): circulant Gaussian row.
// kx_k = 2*pi*k*511/512 (fftfreq with d = 1/511). g is real & symmetric.
// ---------------------------------------------------------------------------
__global__ void k_build_grow(float* __restrict__ grow) {
  int d = blockIdx.x * blockDim.x + threadIdx.x;
  if (d >= HN) return;
  const float TWO_PI = 6.28318530717958647692f;
  const float w1 = TWO_PI * (511.0f / 512.0f);
  const float c = 0.5f * (0.05f * 0.05f) * w1 * w1;
  float s = 1.0f;                                      // k = 0 term
  for (int k = 1; k <= 255; ++k) {
    float F = __expf(-c * (float)(k * k));
    int kd = (k * d) & 511;                            // periodic phase reduction
    s += 2.0f * F * __cosf(TWO_PI * (float)kd * (1.0f / 512.0f));
  }
  s += __expf(-c * 65536.0f) * ((d & 1) ? -1.0f : 1.0f);  // Nyquist (~0)
  grow[d] = s * (1.0f / 512.0f);
}

__global__ void k_build_G(const float* __restrict__ grow, float* __restrict__ G) {
  int i = blockIdx.x * blockDim.x + threadIdx.x;
  if (i >= HN * WN) return;
  int r = i >> 9, c = i & 511;
  G[i] = grow[(r - c) & 511];
}

// ---------------------------------------------------------------------------
// C[512x512] = A[512x512] * B[512x512]  (fp32 WMMA 16x16x4), batched over z.
// aStride/bStride/cStride are per-z element strides (0 => shared operand).
// ---------------------------------------------------------------------------
__global__ __launch_bounds__(256, 2)
void k_gemm_wmma(const float* __restrict__ A, const float* __restrict__ B,
                 float* __restrict__ C, long aStride, long bStride, long cStride) {
  __shared__ float sA[BM * LDA];     // row-major:  sA[m][k]
  __shared__ float sBt[BN * LDBT];   // transposed: sBt[n][k]

  const int tid  = threadIdx.x;
  const int lane = tid & 31;
  const int wave = tid >> 5;                 // 0..7
  const int bz   = blockIdx.z;

  const float* Ab = A + aStride * bz;
  const float* Bb = B + bStride * bz;
  float*       Cb = C + cStride * bz;

  const int m0 = blockIdx.y * BM;            // grid.y = 4
  const int n0 = blockIdx.x * BN;            // grid.x = 8

  v8f acc[4];
  for (int t = 0; t < 4; ++t)
    for (int e = 0; e < 8; ++e) acc[t][e] = 0.0f;

  const int mlane = lane & 15;               // row (A) / col (B) within 16
  const int khalf = (lane >> 4) << 1;        // 0 or 2: K split across lane halves
  const int wrow  = wave * 16 + mlane;       // A-fragment row within block tile

  for (int kb = 0; kb < 512; kb += BK) {
    // Stage A tile: 128x64 floats as 2048 float4 (8 per thread), row-major
    for (int i = tid; i < (BM * BK) / 4; i += 256) {
      int r  = i >> 4;
      int c4 = (i & 15) << 2;
      float4 v = *(const float4*)(Ab + (long)(m0 + r) * 512 + kb + c4);
      float* sp = &sA[r * LDA + c4];
      sp[0] = v.x; sp[1] = v.y; sp[2] = v.z; sp[3] = v.w;
    }
    // Stage B tile: 64x64, coalesced float4 reads along n, transposed scatter
    // into sBt[n][k] (cold-path 4-way bank conflict; hot frag reads conflict-free)
    for (int i = tid; i < (BK * BN) / 4; i += 256) {
      int r  = i >> 4;                       // k-local
      int c4 = (i & 15) << 2;                // n-local
      float4 v = *(const float4*)(Bb + (long)(kb + r) * 512 + n0 + c4);
      sBt[(c4 + 0) * LDBT + r] = v.x;
      sBt[(c4 + 1) * LDBT + r] = v.y;
      sBt[(c4 + 2) * LDBT + r] = v.z;
      sBt[(c4 + 3) * LDBT + r] = v.w;
    }
    __syncthreads();

#pragma unroll 4
    for (int kk = 0; kk < BK; kk += 4) {
      // A fragment 16x4: lanes 0-15 rows, K pair by lane half (aligned b64)
      const v2f a = *(const v2f*)(sA + wrow * LDA + kk + khalf);
#pragma unroll
      for (int t = 0; t < 4; ++t) {
        // B fragment 4x16: lanes 0-15 cols, K pair by lane half (aligned b64)
        const int n = t * 16 + mlane;
        const v2f b = *(const v2f*)(sBt + n * LDBT + kk + khalf);
        acc[t] = __builtin_amdgcn_wmma_f32_16x16x4_f32(
            /*neg_a=*/false, a, /*neg_b=*/false, b,
            /*c_mod=*/(short)0, acc[t], /*reuse_a=*/false, /*reuse_b=*/false);
      }
    }
    __syncthreads();
  }

  // D layout: lane = n + 16*(m/8); VGPR r = m%8
  const int mstore = m0 + wave * 16 + ((lane >> 4) << 3);
#pragma unroll
  for (int t = 0; t < 4; ++t)
#pragma unroll
    for (int r = 0; r < 8; ++r)
      Cb[(long)(mstore + r) * 512 + n0 + t * 16 + mlane] = acc[t][r];
}

// ---------------------------------------------------------------------------
// gIx = dI/dH, gIy = dI/dW (jnp.gradient semantics / spacing), m = lam * grad(I)
// ---------------------------------------------------------------------------
__global__ void k_grad_mom(const float* __restrict__ I, const float* __restrict__ lam,
                           float* __restrict__ gx, float* __restrict__ gy,
                           float* __restrict__ mx, float* __restrict__ my) {
  int i = blockIdx.x * blockDim.x + threadIdx.x;
  if (i >= FIELD) return;
  int w = i & 511;
  int h = (i >> 9) & 511;
  float gxx, gyy;
  if (h == 0)        gxx = (I[i + WN] - I[i]) * INV_H;
  else if (h == 511) gxx = (I[i] - I[i - WN]) * INV_H;
  else               gxx = (I[i + WN] - I[i - WN]) * (0.5f * INV_H);
  if (w == 0)        gyy = (I[i + 1] - I[i]) * INV_H;
  else if (w == 511) gyy = (I[i] - I[i - 1]) * INV_H;
  else               gyy = (I[i + 1] - I[i - 1]) * (0.5f * INV_H);
  float l = lam[i];
  gx[i] = gxx; gy[i] = gyy;
  mx[i] = l * gxx; my[i] = l * gyy;
}

// dI = -(vx*gIx + vy*gIy) (written over gx); px = lam*vx; py = lam*vy
__global__ void k_finalize(const float* __restrict__ vx, const float* __restrict__ vy,
                           float* __restrict__ gx, const float* __restrict__ gy,
                           const float* __restrict__ lam,
                           float* __restrict__ px, float* __restrict__ py) {
  int i = blockIdx.x * blockDim.x + threadIdx.x;
  if (i >= FIELD) return;
  float a = vx[i], b = vy[i], l = lam[i];
  float dI = -(a * gx[i] + b * gy[i]);
  px[i] = l * a;
  py[i] = l * b;
  gx[i] = dI;
}

// dlam = -(d/dH px + d/dW py)
__global__ void k_div(const float* __restrict__ px, const float* __restrict__ py,
                      float* __restrict__ out) {
  int i = blockIdx.x * blockDim.x + threadIdx.x;
  if (i >= FIELD) return;
  int w = i & 511;
  int h = (i >> 9) & 511;
  float dx, dy;
  if (h == 0)        dx = (px[i + WN] - px[i]) * INV_H;
  else if (h == 511) dx = (px[i] - px[i - WN]) * INV_H;
  else               dx = (px[i + WN] - px[i - WN]) * (0.5f * INV_H);
  if (w == 0)        dy = (py[i + 1] - py[i]) * INV_H;
  else if (w == 511) dy = (py[i] - py[i - 1]) * INV_H;
  else               dy = (py[i + 1] - py[i - 1]) * (0.5f * INV_H);
  out[i] = -(dx + dy);
}

// RK4 combine: nxt = (init? cur : nxt) + wAcc*k ; if doTmp: tmp = cur + wTmp*k
__global__ void k_combine(const float* __restrict__ curL, const float* __restrict__ curI,
                          const float* __restrict__ kL, const float* __restrict__ kI,
                          float* __restrict__ nxtL, float* __restrict__ nxtI,
                          float* __restrict__ tmpL, float* __restrict__ tmpI,
                          float wAcc, float wTmp, int flags) {
  int i = blockIdx.x * blockDim.x + threadIdx.x;
  if (i >= FIELD) return;
  float kl = kL[i], ki = kI[i];
  float bl = (flags & 1) ? curL[i] : nxtL[i];
  float bi = (flags & 1) ? curI[i] : nxtI[i];
  nxtL[i] = bl + wAcc * kl;
  nxtI[i] = bi + wAcc * ki;
  if (flags & 2) {
    tmpL[i] = curL[i] + wTmp * kl;
    tmpI[i] = curI[i] + wTmp * ki;
  }
}

// ---------------------------------------------------------------------------
extern "C" void kernel_launch(void* const* d_in, const int* in_sizes, int n_in,
                              void* d_out, int out_size, void* d_ws, size_t ws_size,
                              hipStream_t stream) {
  (void)in_sizes; (void)n_in; (void)out_size; (void)ws_size;
  const float* inI = (const float*)d_in[0];   // I
  const float* inL = (const float*)d_in[1];   // lam

  float* ws   = (float*)d_ws;
  float* G    = ws;                           // 262144 floats
  float* grow = ws + 262144;                  // 512 floats
  float* F0   = ws + 263168;                  // fields, 16B-aligned offsets
  auto FLD = [&](int k) { return F0 + (size_t)k * FIELD; };
  float *curL = FLD(0),  *curI = FLD(1);
  float *nxtL = FLD(2),  *nxtI = FLD(3);
  float *tmpL = FLD(4),  *tmpI = FLD(5);
  float *gX   = FLD(6),  *gY   = FLD(7);      // gIx->kI, gIy->klam
  float *A0   = FLD(8),  *A1   = FLD(9);      // mx/vx, my/vy
  float *B0   = FLD(10), *B1   = FLD(11);     // Tx/px, Ty/py

  k_build_grow<<<2, 256, 0, stream>>>(grow);
  k_build_G<<<(HN * WN) / 256, 256, 0, stream>>>(grow, G);
  hipMemcpyAsync(curI, inI, (size_t)FIELD * sizeof(float), hipMemcpyDeviceToDevice, stream);
  hipMemcpyAsync(curL, inL, (size_t)FIELD * sizeof(float), hipMemcpyDeviceToDevice, stream);

  dim3 gb(FIELD / 256);
  dim3 gg(WN / BN, HN / BM, BATCH);
  const float wA[4] = {DT / 6.0f, DT / 3.0f, DT / 3.0f, DT / 6.0f};
  const float wT[4] = {0.5f * DT, 0.5f * DT, DT, 0.0f};

  for (int step = 0; step < RKSTEPS; ++step) {
    for (int s = 0; s < 4; ++s) {
      const float* iL = (s == 0) ? curL : tmpL;
      const float* iI = (s == 0) ? curI : tmpI;
      k_grad_mom<<<gb, 256, 0, stream>>>(iI, iL, gX, gY, A0, A1);
      // smooth: T = G * m   (G shared: aStride 0)
      k_gemm_wmma<<<gg, 256, 0, stream>>>(G, A0, B0, 0, (long)HW, (long)HW);
      k_gemm_wmma<<<gg, 256, 0, stream>>>(G, A1, B1, 0, (long)HW, (long)HW);
      // v = T * G   (G symmetric)
      k_gemm_wmma<<<gg, 256, 0, stream>>>(B0, G, A0, (long)HW, 0, (long)HW);
      k_gemm_wmma<<<gg, 256, 0, stream>>>(B1, G, A1, (long)HW, 0, (long)HW);
      k_finalize<<<gb, 256, 0, stream>>>(A0, A1, gX, gY, iL, B0, B1);
      k_div<<<gb, 256, 0, stream>>>(B0, B1, gY);
      int flags = (s == 0 ? 1 : 0) | (s < 3 ? 2 : 0);
      k_combine<<<gb, 256, 0, stream>>>(curL, curI, gY, gX, nxtL, nxtI, tmpL, tmpI,
                                        wA[s], wT[s], flags);
    }
    float* t;
    t = curL; curL = nxtL; nxtL = t;
    t = curI; curI = nxtI; nxtI = t;
  }
  hipMemcpyAsync(d_out, curI, (size_t)FIELD * sizeof(float), hipMemcpyDeviceToDevice, stream);
}